// Mind3D_88931592831196
// MI455X (gfx1250) — compile-verified
//
#include <hip/hip_runtime.h>
#include <math.h>

// ---------------------------------------------------------------------------
// MIND-3D descriptor, fused two-pass implementation for gfx1250 (MI455X).
//
//  * diff_c(q) = x[q+off1_c] - x[q+off2_c]   (one-hot dilated conv == shifts)
//  * 5x5x5 box pooling as banded matmuls via V_WMMA_F32_16X16X4_F32:
//      C2(16x16) = SyT(16x20) x Dsq(20x20) x Sx(20x16)
//    band matrices built per-lane from clamped global indices (encodes the
//    replication-pad edge multiplicities exactly); hoisted out of all loops.
//  * x halo staged per z-slice into an LDS ring (5 slices of 24x24), with
//    conv zero-padding and replication clamping baked in at staging time;
//    staging uses gfx1250 global_load_async_to_lds + s_wait_asynccnt when the
//    toolchain exposes the builtins (fallback: load + ds_store).
//  * z direction: 5-tap weighted accumulation of pooled slices into LDS.
//  * mind/mind_var is scale invariant => 1/125 pool normalization dropped.
//  * pass A reduces mean(mind_var) into ws[0]; pass B recomputes + exp +
//    nontemporal-stores (x is 8MB L2-resident; recompute beats spilling the
//    105MB ssd tensor to HBM; NT stores keep the streamed output out of L2).
// ---------------------------------------------------------------------------

typedef __attribute__((ext_vector_type(2))) float v2f;
typedef __attribute__((ext_vector_type(8))) float v8f;

#define TZ 3           // output z-slices per block
#define NOUT 130       // output spatial extent per dim

#if defined(__has_builtin)
#if __has_builtin(__builtin_amdgcn_global_load_async_to_lds_b32)
#define ASYNC_LDS 1
#endif
#endif
#ifndef ASYNC_LDS
#define ASYNC_LDS 0
#endif

typedef __attribute__((address_space(1))) int as1_int;  // global
typedef __attribute__((address_space(3))) int as3_int;  // LDS

__device__ __forceinline__ void async_wait_all() {
#if defined(__has_builtin) && __has_builtin(__builtin_amdgcn_s_wait_asynccnt)
  __builtin_amdgcn_s_wait_asynccnt(0);
#elif ASYNC_LDS
  asm volatile("s_wait_asynccnt 0x0" ::: "memory");
#endif
}

// per-channel shift offsets (z,y,x), off = 2*kernel_idx - 1, idx in {0,1,2}
__constant__ int g_o1[12][3] = {
  { 1, 1,-1},{ 1,-1, 1},{ 1,-1, 1},{ 1, 1, 3},{ 1, 1, 3},{ 3, 1, 1},
  { 3, 1, 1},{ 3, 1, 1},{ 1, 3, 1},{ 1, 3, 1},{ 1, 3, 1},{ 1, 3, 1}};
__constant__ int g_o2[12][3] = {
  {-1, 1, 1},{-1, 1, 1},{ 1, 1,-1},{-1, 1, 1},{ 1,-1, 1},{ 1, 1,-1},
  { 1,-1, 1},{ 1, 1, 3},{-1, 1, 1},{ 1, 1,-1},{ 1, 1, 3},{ 3, 1, 1}};

__device__ __forceinline__ int iclamp(int v, int lo, int hi) {
  return v < lo ? lo : (v > hi ? hi : v);
}

// # of pooling windows of output voxel p that land on diff index q
// (replication pad radius 2, window 5, diff extent [0,130))
__device__ __forceinline__ float poolW(int p, int q) {
  int cnt = 0;
#pragma unroll
  for (int w = 0; w < 5; ++w) {
    int t = iclamp(p + w - 2, 0, NOUT - 1);
    cnt += (t == q) ? 1 : 0;
  }
  return (float)cnt;
}

__device__ __forceinline__ v8f zero8() {
  v8f z = {0.f, 0.f, 0.f, 0.f, 0.f, 0.f, 0.f, 0.f};
  return z;
}

__device__ __forceinline__ int ring(int t) { return ((t % 5) + 5) % 5; }

template <int WRITE>
__global__ __launch_bounds__(128)
void mind_tile(const float* __restrict__ X, float* __restrict__ out,
               float* __restrict__ wsum) {
  __shared__ float s_ssd[12 * TZ * 256];  // unnormalized ssd tile (36 KB)
  __shared__ float s_x[5][24 * 24];       // x slice ring, conv frame (11.25 KB)
  __shared__ float s_scr[4][16 * 20];     // per-wave C->A layout scratch (5 KB)
  __shared__ float s_red;

  const int tid = (int)threadIdx.x;
  const int lane = tid & 31;
  const int wv = tid >> 5;      // 4 waves, 3 channels each
  const int hf = lane >> 4;     // lane half (WMMA operand layout)
  const int n16 = lane & 15;

  const int px0 = (int)blockIdx.x * 16;
  const int py0 = (int)blockIdx.y * 16;
  const int z0 = (int)blockIdx.z * TZ;

  for (int i = tid; i < 12 * TZ * 256; i += 128) s_ssd[i] = 0.0f;
  if (tid == 0) s_red = 0.0f;

  // hoisted band matrices: Sy^T chunks (A operand) and Sx chunks (B operand)
  v2f ay[5], bx[5];
#pragma unroll
  for (int k = 0; k < 5; ++k) {
    const int r0 = 4 * k + 2 * hf;
    ay[k][0] = poolW(py0 + n16, py0 - 2 + r0);
    ay[k][1] = poolW(py0 + n16, py0 - 2 + r0 + 1);
    bx[k][0] = poolW(px0 + n16, px0 - 2 + r0);
    bx[k][1] = poolW(px0 + n16, px0 - 2 + r0 + 1);
  }

  // Stage x slice t (conv-frame z) into the LDS ring. Value semantics:
  // 0 outside [0,132)^3 (conv zero-pad), else X[clamp(t-2,0,127)] (edge pad).
  auto stage = [&](int t) {
    const int slot = ring(t);
    const bool vz = ((unsigned)t < 132u);
    const int iz = iclamp(t - 2, 0, 127);
    for (int e = tid; e < 576; e += 128) {
      const int ly = e / 24, lx = e % 24;
      const int ty = py0 - 3 + ly, tx = px0 - 3 + lx;
      const bool v = vz & ((unsigned)ty < 132u) & ((unsigned)tx < 132u);
      if (v) {
        const int iy = iclamp(ty - 2, 0, 127);
        const int ix = iclamp(tx - 2, 0, 127);
        float* gp = const_cast<float*>(&X[((size_t)iz * 128 + iy) * 128 + ix]);
#if ASYNC_LDS
        __builtin_amdgcn_global_load_async_to_lds_b32(
            (as1_int*)gp, (as3_int*)&s_x[slot][e], /*offset=*/0, /*cpol=*/0);
#else
        s_x[slot][e] = *gp;
#endif
      } else {
        s_x[slot][e] = 0.0f;
      }
    }
  };

  // prologue: slices t = z0-3 .. z0+1 (first qz = z0-2 needs t in {qz-1,qz+1,qz+3})
  for (int t = z0 - 3; t <= z0 + 1; ++t) stage(t);
  async_wait_all();
  __syncthreads();

  for (int s = 0; s < TZ + 4; ++s) {
    const int qz = z0 - 2 + s;            // diff z-slice (block-uniform)
    if (s > 0) {
      stage(qz + 3);                      // evicts t = qz-2, unused since s-1
      async_wait_all();
      __syncthreads();
    }
    if (qz >= 0 && qz <= NOUT - 1) {
      float wzv[TZ];
#pragma unroll
      for (int pzl = 0; pzl < TZ; ++pzl) wzv[pzl] = poolW(z0 + pzl, qz);

      for (int ci = 0; ci < 3; ++ci) {
        const int c = wv * 3 + ci;        // wave-uniform channel
        const float* xz1 = s_x[ring(qz + g_o1[c][0])];
        const float* xz2 = s_x[ring(qz + g_o2[c][0])];
        const int o1 = g_o1[c][1] * 24 + g_o1[c][2];  // (y,x) offset, flat
        const int o2 = g_o2[c][1] * 24 + g_o2[c][2];

        // ---- pass Y: C1(16x20) = SyT(16x20) x Dsq(20x20), K chunks of 4 ----
        v8f c1a = zero8();  // dsq columns x' = 0..15
        v8f c1b = zero8();  // dsq columns x' = 16..19 (in output cols 0..3)
#pragma unroll
        for (int k = 0; k < 5; ++k) {
          const int r0 = 4 * k + 2 * hf;  // patch-local K row for this lane
          const int base0 = (r0 + 1) * 24 + (n16 + 1);
          v2f b0, b1;
          {
            const float g0 = xz1[base0 + o1] - xz2[base0 + o2];
            const float g1 = xz1[base0 + 24 + o1] - xz2[base0 + 24 + o2];
            b0[0] = g0 * g0;
            b0[1] = g1 * g1;
          }
          {
            const int base1 = (r0 + 1) * 24 + (16 + (n16 & 3) + 1);
            const float m1 = (n16 < 4) ? 1.0f : 0.0f;
            const float g0 = xz1[base1 + o1] - xz2[base1 + o2];
            const float g1 = xz1[base1 + 24 + o1] - xz2[base1 + 24 + o2];
            b1[0] = m1 * g0 * g0;
            b1[1] = m1 * g1 * g1;
          }
          c1a = __builtin_amdgcn_wmma_f32_16x16x4_f32(
              false, ay[k], false, b0, (short)0, c1a, false, false);
          c1b = __builtin_amdgcn_wmma_f32_16x16x4_f32(
              false, ay[k], false, b1, (short)0, c1b, false, false);
        }

        // ---- C-layout -> A-layout transpose through per-wave LDS scratch ----
        float* scr = &s_scr[wv][0];
#pragma unroll
        for (int r = 0; r < 8; ++r) {
          const int y = r + 8 * hf;
          scr[y * 20 + n16] = c1a[r];
          if (n16 < 4) scr[y * 20 + 16 + n16] = c1b[r];
        }
        __syncthreads();

        // ---- pass X: C2(16x16) = C1(16x20) x Sx(20x16) ----
        v8f c2 = zero8();
#pragma unroll
        for (int k = 0; k < 5; ++k) {
          const int r0 = 4 * k + 2 * hf;
          v2f a2;
          a2[0] = scr[n16 * 20 + r0];
          a2[1] = scr[n16 * 20 + r0 + 1];
          c2 = __builtin_amdgcn_wmma_f32_16x16x4_f32(
              false, a2, false, bx[k], (short)0, c2, false, false);
        }
        __syncthreads();  // scratch + s_x reads done; safe for next writes

        // ---- z accumulation with pooling weights (wave owns channel c) ----
#pragma unroll
        for (int pzl = 0; pzl < TZ; ++pzl) {
          if (wzv[pzl] != 0.0f) {
#pragma unroll
            for (int r = 0; r < 8; ++r) {
              const int y = r + 8 * hf;
              s_ssd[((c * TZ + pzl) * 16 + y) * 16 + n16] += wzv[pzl] * c2[r];
            }
          }
        }
      }  // ci
    }    // qz valid
  }      // s
  __syncthreads();

  // ---- epilogue: channel min/mean per voxel ----
  float acc = 0.0f;
  float lo = 0.0f, hi = 0.0f;
  if (WRITE) {
    const float mvm = wsum[0] * (1.0f / 2197000.0f);  // / 130^3
    lo = mvm * 0.001f;
    hi = mvm * 1000.0f;
  }
  for (int v = tid; v < 16 * 16 * TZ; v += 128) {
    const int x = v & 15, y = (v >> 4) & 15, pzl = v >> 8;
    const int px = px0 + x, py = py0 + y, pz = z0 + pzl;
    if (px < NOUT && py < NOUT && pz < NOUT) {
      float sv[12];
      float smin = 3.4e38f, ssum = 0.0f;
#pragma unroll
      for (int c = 0; c < 12; ++c) {
        sv[c] = s_ssd[((c * TZ + pzl) * 16 + y) * 16 + x];
        smin = fminf(smin, sv[c]);
        ssum += sv[c];
      }
      const float mv = ssum * (1.0f / 12.0f) - smin;
      if (WRITE) {
        const float mvc = fminf(fmaxf(mv, lo), hi);
        const float inv = 1.0f / mvc;
#pragma unroll
        for (int c = 0; c < 12; ++c) {
          __builtin_nontemporal_store(
              expf(-(sv[c] - smin) * inv),
              &out[(((size_t)c * NOUT + pz) * NOUT + py) * NOUT + px]);
        }
      } else {
        acc += mv;
      }
    }
  }
  if (!WRITE) {
    atomicAdd(&s_red, acc);
    __syncthreads();
    if (tid == 0) atomicAdd(wsum, s_red);
  }
}

__global__ void mind_zero(float* w) {
  if (threadIdx.x == 0) w[0] = 0.0f;
}

extern "C" void kernel_launch(void* const* d_in, const int* in_sizes, int n_in,
                              void* d_out, int out_size, void* d_ws,
                              size_t ws_size, hipStream_t stream) {
  const float* X = (const float*)d_in[0];
  float* out = (float*)d_out;
  float* wsum = (float*)d_ws;

  dim3 grid((NOUT + 15) / 16, (NOUT + 15) / 16, (NOUT + TZ - 1) / TZ);
  dim3 block(128);

  mind_zero<<<dim3(1), dim3(32), 0, stream>>>(wsum);
  mind_tile<0><<<grid, block, 0, stream>>>(X, out, wsum);  // mv_mean reduction
  mind_tile<1><<<grid, block, 0, stream>>>(X, out, wsum);  // final output
}